// CameraPoseModel_21268678050229
// MI455X (gfx1250) — compile-verified
//
#include <hip/hip_runtime.h>
#include <math.h>

typedef __attribute__((ext_vector_type(2))) float v2f;
typedef __attribute__((ext_vector_type(8))) float v8f;

// Transposed batched 4x4 matmul via V_WMMA_F32_16X16X4_F32:
//   A (16x4)  = 4 stacked P^T   (rows 4q+i = column i of pose q)   [global]
//   B (4x16)  = 4 c2w^T side by side (cols 4q+j = row j of c2w_q)  [LDS]
//   D (16x16) = diagonal block (i,j) = out_q[j][i]  ->  each valid lane holds
//               one contiguous ROW of out_q  ->  single b128 store.
//
// Phase 1: one lane = one pose, full c2w built straight-line, parked in LDS.
// Phase 2: per group: 2x b32 (A), 1x ds_load_b64 (B), 1 WMMA, 1x b128 store.
// Fast path (whole 32-pose range in bounds) has no per-group guards so the
// scheduler can clause all loads across groups.

constexpr int GROUPS = 8;              // 8 WMMAs -> 32 poses per wave
constexpr int WAVES_PER_BLOCK = 8;     // 256 threads

__global__ __launch_bounds__(256) void camera_pose_wmma_kernel(
    const float* __restrict__ poses,
    const float* __restrict__ r,
    const float* __restrict__ t,
    float* __restrict__ out,
    int N)
{
    __shared__ float lds[WAVES_PER_BLOCK * 32 * 16];   // 16 KB: one c2w per pose

    const int lane = threadIdx.x & 31;
    const int wIB  = threadIdx.x >> 5;
    // force wave-uniform values into SGPRs so guards stay scalar (EXEC untouched)
    const int wave = __builtin_amdgcn_readfirstlane(blockIdx.x * WAVES_PER_BLOCK + wIB);
    const int base = wave * (4 * GROUPS);
    if (base >= N) return;                    // scalar branch

    // ---------- phase 1: lane builds c2w for pose (base+lane), no selects ----------
    {
        const int p  = base + lane;
        const int pc = (p < N) ? p : (N - 1);             // clamp (tail safety)
        const float rx = r[3*pc+0], ry = r[3*pc+1], rz = r[3*pc+2];
        const float tx = t[3*pc+0], ty = t[3*pc+1], tz = t[3*pc+2];

        // R = I + sA*K + cB*K^2,  K^2 = r r^T - n2*I
        const float n2  = fmaf(rx, rx, fmaf(ry, ry, rz * rz));
        const float th  = __builtin_amdgcn_sqrtf(n2) + 1e-15f;
        const float inv = __builtin_amdgcn_rcpf(th);
        const float sA  = __sinf(th) * inv;
        const float cB  = (1.0f - __cosf(th)) * (inv * inv);
        const float dg  = fmaf(-cB, n2, 1.0f);            // 1 - cB*n2

        const float cbx = cB * rx, cby = cB * ry, cbz = cB * rz;
        const float sax = sA * rx, say = sA * ry, saz = sA * rz;

        float4* row = (float4*)&lds[(wIB * 32 + lane) * 16];
        row[0] = make_float4(fmaf(cbx, rx,  dg), fmaf(cbx, ry, -saz), fmaf(cbx, rz,  say), tx);
        row[1] = make_float4(fmaf(cby, rx,  saz), fmaf(cby, ry,  dg), fmaf(cby, rz, -sax), ty);
        row[2] = make_float4(fmaf(cbz, rx, -say), fmaf(cbz, ry,  sax), fmaf(cbz, rz,  dg), tz);
        row[3] = make_float4(0.0f, 0.0f, 0.0f, 1.0f);
    }
    // same-wave producer/consumer: DS ops are in-order; compiler inserts dscnt wait.

    // ---------- phase 2: 8 block-diagonal WMMAs (transposed) ----------
    const int half = lane >> 4;               // 0 -> K in {0,1}; 1 -> K in {2,3}
    const int m    = lane & 15;               // A-row / B,D-column index
    const int q    = m >> 2;                  // pose within group (0..3)
    const int i    = m & 3;                   // A: column-of-P index; B/D: out-row index
    const bool hi    = (q & 1);               // odd pose-in-group -> D VGPRs 4..7
    const bool valid = (half == (m >> 3));    // lanes holding diagonal blocks

    // loop-invariant per-lane pointers; bodies use immediate offsets (+64 floats/group)
    const float* ap = poses + 16 * (size_t)(base + q) + 8 * half + i;  // P[2h][i], P[2h+1][i]
    const float* bp = &lds[(wIB * 32 + q) * 16 + i * 4 + half * 2];    // C[i][2h], C[i][2h+1]
    float*       op = out   + 16 * (size_t)(base + q) + 4 * i;         // row i of out pose

    auto do_group = [&](int g) {
        v2f a;
        a.x = ap[64 * g];                     // A[m][2*half]   = P_q[2h  ][i]
        a.y = ap[64 * g + 4];                 // A[m][2*half+1] = P_q[2h+1][i]
        const v2f b = *(const v2f*)(bp + 64 * g);       // B[2h..2h+1][m] = C_q[i][2h..2h+1]

        v8f c = {};
        c = __builtin_amdgcn_wmma_f32_16x16x4_f32(
            false, a, false, b, (short)0, c, false, false);

        const float o0 = hi ? c[4] : c[0];
        const float o1 = hi ? c[5] : c[1];
        const float o2 = hi ? c[6] : c[2];
        const float o3 = hi ? c[7] : c[3];
        if (valid) {
            *(float4*)(op + 64 * g) = make_float4(o0, o1, o2, o3);
        }
    };

    if (base + 4 * GROUPS <= N) {
        // fast path: no per-group guards -> loads can clause across groups
#pragma unroll
        for (int g = 0; g < GROUPS; ++g) do_group(g);
    } else {
        // ragged path: per-group scalar guards
#pragma unroll
        for (int g = 0; g < GROUPS; ++g) {
            if (base + 4 * g + 4 <= N) do_group(g);
        }
        // tail: one partial group (only when N % 4 != 0)
        const int tg = (N - base) >> 2;
        if (tg < GROUPS) {
            const int tbase = base + 4 * tg;
            if (tbase < N) {
                const int p  = tbase + q;
                const int pc = (p < N) ? p : (N - 1);     // clamp addresses
                const float* apc = poses + 16 * (size_t)pc + 8 * half + i;
                v2f a;
                a.x = apc[0];
                a.y = apc[4];
                const v2f b = *(const v2f*)(bp + 64 * tg);  // clamped-but-valid c2w

                v8f c = {};
                c = __builtin_amdgcn_wmma_f32_16x16x4_f32(
                    false, a, false, b, (short)0, c, false, false);

                const float o0 = hi ? c[4] : c[0];
                const float o1 = hi ? c[5] : c[1];
                const float o2 = hi ? c[6] : c[2];
                const float o3 = hi ? c[7] : c[3];
                if (valid && (p < N)) {
                    *(float4*)(out + 16 * (size_t)p + 4 * i) = make_float4(o0, o1, o2, o3);
                }
            }
        }
    }
}

extern "C" void kernel_launch(void* const* d_in, const int* in_sizes, int n_in,
                              void* d_out, int out_size, void* d_ws, size_t ws_size,
                              hipStream_t stream) {
    (void)n_in; (void)out_size; (void)d_ws; (void)ws_size;
    const float* poses = (const float*)d_in[0];
    const float* r     = (const float*)d_in[1];
    const float* t     = (const float*)d_in[2];
    float* out         = (float*)d_out;

    const int N = in_sizes[0] / 16;                       // number of 4x4 poses
    const int poses_per_wave = 4 * GROUPS;                // 32
    const long long waves = ((long long)N + poses_per_wave - 1) / poses_per_wave;
    const int threads = 32 * WAVES_PER_BLOCK;             // 256
    const int blocks  = (int)((waves + WAVES_PER_BLOCK - 1) / WAVES_PER_BLOCK);

    camera_pose_wmma_kernel<<<blocks, threads, 0, stream>>>(poses, r, t, out, N);
}